// GAT_68264210202658
// MI455X (gfx1250) — compile-verified
//
#include <hip/hip_runtime.h>
#include <math.h>

#define NN   50000
#define EE   800000
#define EP   (EE + NN)        // edges + self loops = 850000
#define INC  128
#define HIDC 32
#define OUTC 16
#define HEADS 8
#define HID  (HEADS * HIDC)   // 256
#define MTILES (NN / 16)      // 3125 exactly

typedef __attribute__((ext_vector_type(2))) float v2f;
typedef __attribute__((ext_vector_type(8))) float v8f;

// ---- order-preserving float<->int encoding for atomicMax on floats ----
__device__ __forceinline__ int enc_f32(float f) {
    int i = __float_as_int(f);
    return i >= 0 ? i : (i ^ 0x7FFFFFFF);
}
__device__ __forceinline__ float dec_f32(int i) {
    return __int_as_float(i >= 0 ? i : (i ^ 0x7FFFFFFF));
}
#define ENC_NEG_INF ((int)0x807FFFFF)  // enc(-inf)

__device__ __forceinline__ float lrelu(float v) { return v > 0.f ? v : 0.2f * v; }

// ------------------------- fills -------------------------
__global__ void gat_fill_f32(float* __restrict__ p, long n, float v) {
    long i = (long)blockIdx.x * 256 + threadIdx.x;
    if (i < n) p[i] = v;
}
__global__ void gat_fill_i32(int* __restrict__ p, long n, int v) {
    long i = (long)blockIdx.x * 256 + threadIdx.x;
    if (i < n) p[i] = v;
}

// ------------------- GEMM1: h = x @ W1 (fp32 WMMA, LDS-staged A) -------------------
// grid 3125, block 256 (8 waves). Block stages its 16x128 x-tile in LDS via the
// gfx1250 async-to-LDS path; each wave computes TWO 16x16 output tiles
// (colTile = wave, wave+8), sharing one A fragment per k-step for two WMMAs.
__global__ void gat_gemm1(const float* __restrict__ x, const float* __restrict__ W,
                          float* __restrict__ h) {
    __shared__ __align__(16) float xt[16 * INC];   // 8 KB tile

    const int t    = threadIdx.x;
    const int wave = t >> 5;
    const int lane = t & 31;
    const int r    = lane & 15;
    const int kh   = lane >> 4;            // 0 or 1: which K-half of the 16x4 fragment
    const int rowTile = blockIdx.x;

    // --- async copy x tile (16 rows x 128 f32 = 8192 B) into LDS: 2 x b128/thread ---
    {
        unsigned long long g0 = (unsigned long long)(uintptr_t)(x)
                              + (unsigned long long)rowTile * (16 * INC * 4)
                              + (unsigned long long)t * 16;
        unsigned int l0 = (unsigned int)(uintptr_t)xt + (unsigned int)t * 16;
        asm volatile("global_load_async_to_lds_b128 %0, %1, off"
                     :: "v"(l0), "v"(g0) : "memory");
        unsigned long long g1 = g0 + 4096;
        unsigned int l1 = l0 + 4096;
        asm volatile("global_load_async_to_lds_b128 %0, %1, off"
                     :: "v"(l1), "v"(g1) : "memory");
        asm volatile("s_wait_asynccnt 0x0" ::: "memory");
    }
    __syncthreads();

    const int colT0 = wave;          // 0..7
    const int colT1 = wave + 8;      // 8..15

    v8f c0 = {0.f, 0.f, 0.f, 0.f, 0.f, 0.f, 0.f, 0.f};
    v8f c1 = {0.f, 0.f, 0.f, 0.f, 0.f, 0.f, 0.f, 0.f};

    const float* a_lds = xt + (size_t)r * INC + kh * 2;          // A[m][k+2kh ..]
    const float* wc0 = W + colT0 * 16 + r + (size_t)(kh * 2) * HID;
    const float* wc1 = W + colT1 * 16 + r + (size_t)(kh * 2) * HID;

    #pragma unroll 4
    for (int k = 0; k < INC; k += 4) {
        v2f a = *(const v2f*)(a_lds + k);                        // ds_load_b64
        v2f b0, b1;
        b0.x = wc0[(size_t)k * HID];
        b0.y = wc0[(size_t)k * HID + HID];
        b1.x = wc1[(size_t)k * HID];
        b1.y = wc1[(size_t)k * HID + HID];
        c0 = __builtin_amdgcn_wmma_f32_16x16x4_f32(false, a, false, b0,
                                                   (short)0, c0, false, false);
        c1 = __builtin_amdgcn_wmma_f32_16x16x4_f32(false, a, false, b1,
                                                   (short)0, c1, false, false);
    }
    // C/D layout: VGPR v -> row v + 8*kh, col = lane&15
    float* hp0 = h + (size_t)(rowTile * 16 + kh * 8) * HID + colT0 * 16 + r;
    float* hp1 = h + (size_t)(rowTile * 16 + kh * 8) * HID + colT1 * 16 + r;
    #pragma unroll
    for (int v = 0; v < 8; ++v) {
        hp0[(size_t)v * HID] = c0[v];
        hp1[(size_t)v * HID] = c1[v];
    }
}

// ------------------- GEMM2: z = h2 @ W2 (fp32 WMMA) -------------------
// grid 391, block 256: wave -> one 16x16 tile (only 1 col tile since OUTC=16)
__global__ void gat_gemm2(const float* __restrict__ h2, const float* __restrict__ W2,
                          float* __restrict__ z) {
    const int wave = threadIdx.x >> 5;
    const int lane = threadIdx.x & 31;
    const int rowTile = blockIdx.x * 8 + wave;
    if (rowTile >= MTILES) return;         // wave-uniform, EXEC stays all-1 inside
    const int r  = lane & 15;
    const int kh = lane >> 4;
    const int m  = rowTile * 16 + r;

    v8f c = {0.f, 0.f, 0.f, 0.f, 0.f, 0.f, 0.f, 0.f};
    const float* hr = h2 + (size_t)m * HID + kh * 2;
    const float* wc = W2 + r + (size_t)(kh * 2) * OUTC;

    #pragma unroll 4
    for (int k = 0; k < HID; k += 4) {
        v2f a = *(const v2f*)(hr + k);
        v2f b;
        b.x = wc[(size_t)k * OUTC];
        b.y = wc[(size_t)k * OUTC + OUTC];
        c = __builtin_amdgcn_wmma_f32_16x16x4_f32(false, a, false, b,
                                                  (short)0, c, false, false);
    }
    float* zp = z + (size_t)(rowTile * 16 + kh * 8) * OUTC + r;
    #pragma unroll
    for (int v = 0; v < 8; ++v) zp[(size_t)v * OUTC] = c[v];
}

// --------- per-node attention coefficients, layer 1 (wave per head) ---------
__global__ void gat_attn1(const float* __restrict__ h, const float* __restrict__ asrc,
                          const float* __restrict__ adst,
                          float* __restrict__ as1, float* __restrict__ ad1) {
    const int n    = blockIdx.x;
    const int hd   = threadIdx.x >> 5;
    const int lane = threadIdx.x & 31;
    float p = h[(size_t)n * HID + hd * HIDC + lane];
    float s = p * asrc[hd * HIDC + lane];
    float d = p * adst[hd * HIDC + lane];
    #pragma unroll
    for (int off = 16; off; off >>= 1) {
        s += __shfl_xor(s, off, 32);
        d += __shfl_xor(d, off, 32);
    }
    if (lane == 0) {
        as1[n * HEADS + hd] = s;
        ad1[n * HEADS + hd] = d;
    }
}

// --------- edge pass A (layer 1): segment max per (dst, head) ---------
__global__ void gat_emax1(const int* __restrict__ ei, const float* __restrict__ as1,
                          const float* __restrict__ ad1, int* __restrict__ m1) {
    long e = (long)blockIdx.x * 256 + threadIdx.x;
    if (e >= EP) return;
    int s, d;
    if (e < EE) { s = ei[e]; d = ei[EE + e]; }
    else        { s = d = (int)(e - EE); }
    #pragma unroll
    for (int hd = 0; hd < HEADS; ++hd) {
        float v = lrelu(as1[s * HEADS + hd] + ad1[d * HEADS + hd]);
        atomicMax(&m1[d * HEADS + hd], enc_f32(v));
    }
}

// --------- edge pass B (layer 1): exp, denom, aggregate (wave per edge) ---------
__global__ void gat_esoft1(const int* __restrict__ ei, const float* __restrict__ as1,
                           const float* __restrict__ ad1, const int* __restrict__ m1,
                           const float* __restrict__ h, float* __restrict__ den1,
                           float* __restrict__ acc1) {
    const int wave = threadIdx.x >> 5;
    const int lane = threadIdx.x & 31;
    long e = (long)blockIdx.x * 8 + wave;
    if (e >= EP) return;                 // wave-uniform
    int s, d;
    if (e < EE) { s = ei[e]; d = ei[EE + e]; }
    else        { s = d = (int)(e - EE); }

    float exv = 0.f;
    if (lane < HEADS) {
        float v = lrelu(as1[s * HEADS + lane] + ad1[d * HEADS + lane]);
        exv = expf(v - dec_f32(m1[d * HEADS + lane]));
        atomicAdd(&den1[d * HEADS + lane], exv);
    }
    #pragma unroll
    for (int hd = 0; hd < HEADS; ++hd) {
        float ex = __shfl(exv, hd, 32);
        float hv = h[(size_t)s * HID + hd * HIDC + lane];
        atomicAdd(&acc1[(size_t)d * HID + hd * HIDC + lane], ex * hv);
    }
}

// --------- layer-1 epilogue: normalize + bias + ELU -> h2 (overwrites h buf) ---------
__global__ void gat_act1(const float* __restrict__ acc1, const float* __restrict__ den1,
                         const float* __restrict__ b1, float* __restrict__ h2) {
    long i = (long)blockIdx.x * 256 + threadIdx.x;
    if (i >= (long)NN * HID) return;
    int n = (int)(i >> 8);
    int col = (int)(i & 255);
    int hd = col >> 5;
    float v = acc1[i] / den1[n * HEADS + hd] + b1[col];
    h2[i] = v > 0.f ? v : (expf(v) - 1.f);   // ELU(alpha=1)
}

// --------- per-node attention coefficients, layer 2 ---------
__global__ void gat_attn2(const float* __restrict__ z, const float* __restrict__ asrc2,
                          const float* __restrict__ adst2,
                          float* __restrict__ as2, float* __restrict__ ad2) {
    int n = blockIdx.x * 256 + threadIdx.x;
    if (n >= NN) return;
    float s = 0.f, d = 0.f;
    #pragma unroll
    for (int c = 0; c < OUTC; ++c) {
        float zz = z[(size_t)n * OUTC + c];
        s += zz * asrc2[c];
        d += zz * adst2[c];
    }
    as2[n] = s;
    ad2[n] = d;
}

// --------- edge pass A (layer 2) ---------
__global__ void gat_emax2(const int* __restrict__ ei, const float* __restrict__ as2,
                          const float* __restrict__ ad2, int* __restrict__ m2) {
    long e = (long)blockIdx.x * 256 + threadIdx.x;
    if (e >= EP) return;
    int s, d;
    if (e < EE) { s = ei[e]; d = ei[EE + e]; }
    else        { s = d = (int)(e - EE); }
    float v = lrelu(as2[s] + ad2[d]);
    atomicMax(&m2[d], enc_f32(v));
}

// --------- edge pass B (layer 2): thread per (edge, channel) ---------
__global__ void gat_esoft2(const int* __restrict__ ei, const float* __restrict__ as2,
                           const float* __restrict__ ad2, const int* __restrict__ m2,
                           const float* __restrict__ z, float* __restrict__ den2,
                           float* __restrict__ acc2) {
    long t = (long)blockIdx.x * 256 + threadIdx.x;
    long e = t >> 4;
    int  c = (int)(t & 15);
    if (e >= EP) return;
    int s, d;
    if (e < EE) { s = ei[e]; d = ei[EE + e]; }
    else        { s = d = (int)(e - EE); }
    float v  = lrelu(as2[s] + ad2[d]);
    float ex = expf(v - dec_f32(m2[d]));
    if (c == 0) atomicAdd(&den2[d], ex);
    atomicAdd(&acc2[(size_t)d * OUTC + c], ex * z[(size_t)s * OUTC + c]);
}

// --------- final: normalize + bias -> d_out ---------
__global__ void gat_final(const float* __restrict__ acc2, const float* __restrict__ den2,
                          const float* __restrict__ b2, float* __restrict__ out) {
    long i = (long)blockIdx.x * 256 + threadIdx.x;
    if (i >= (long)NN * OUTC) return;
    int n = (int)(i >> 4);
    int c = (int)(i & 15);
    out[i] = acc2[i] / den2[n] + b2[c];
}

extern "C" void kernel_launch(void* const* d_in, const int* in_sizes, int n_in,
                              void* d_out, int out_size, void* d_ws, size_t ws_size,
                              hipStream_t stream) {
    const float* x    = (const float*)d_in[0];
    const int*   ei   = (const int*)d_in[1];     // edge_index [2, E]
    const float* W1   = (const float*)d_in[2];
    const float* asc1 = (const float*)d_in[3];
    const float* ads1 = (const float*)d_in[4];
    const float* b1   = (const float*)d_in[5];
    const float* W2   = (const float*)d_in[6];
    const float* asc2 = (const float*)d_in[7];
    const float* ads2 = (const float*)d_in[8];
    const float* b2   = (const float*)d_in[9];
    float* out = (float*)d_out;

    // workspace layout (floats)
    float* ws   = (float*)d_ws;
    float* h    = ws;                                 // N*256 (h1, then h2 in-place)
    float* acc1 = h + (size_t)NN * HID;               // N*256
    float* as1  = acc1 + (size_t)NN * HID;            // N*8
    float* ad1  = as1 + (size_t)NN * HEADS;           // N*8
    float* den1 = ad1 + (size_t)NN * HEADS;           // N*8
    int*   m1   = (int*)(den1 + (size_t)NN * HEADS);  // N*8 (encoded)
    float* z    = (float*)(m1 + (size_t)NN * HEADS);  // N*16
    float* acc2 = z + (size_t)NN * OUTC;              // N*16
    float* as2  = acc2 + (size_t)NN * OUTC;           // N
    float* ad2  = as2 + NN;                           // N
    float* den2 = ad2 + NN;                           // N
    int*   m2   = (int*)(den2 + NN);                  // N (encoded)

    const long nHID  = (long)NN * HID;
    const long nH8   = (long)NN * HEADS;
    const long nO    = (long)NN * OUTC;

    // init accumulators / maxima
    gat_fill_f32<<<(nHID + 255) / 256, 256, 0, stream>>>(acc1, nHID, 0.f);
    gat_fill_f32<<<(nH8 + 255) / 256, 256, 0, stream>>>(den1, nH8, 0.f);
    gat_fill_i32<<<(nH8 + 255) / 256, 256, 0, stream>>>(m1, nH8, ENC_NEG_INF);
    gat_fill_f32<<<(nO + 255) / 256, 256, 0, stream>>>(acc2, nO, 0.f);
    gat_fill_f32<<<(NN + 255) / 256, 256, 0, stream>>>(den2, NN, 0.f);
    gat_fill_i32<<<(NN + 255) / 256, 256, 0, stream>>>(m2, NN, ENC_NEG_INF);

    // layer 1
    gat_gemm1<<<MTILES, 256, 0, stream>>>(x, W1, h);
    gat_attn1<<<NN, 256, 0, stream>>>(h, asc1, ads1, as1, ad1);
    gat_emax1<<<(EP + 255) / 256, 256, 0, stream>>>(ei, as1, ad1, m1);
    gat_esoft1<<<(EP + 7) / 8, 256, 0, stream>>>(ei, as1, ad1, m1, h, den1, acc1);
    gat_act1<<<(nHID + 255) / 256, 256, 0, stream>>>(acc1, den1, b1, h);  // h := h2

    // layer 2
    gat_gemm2<<<(MTILES + 7) / 8, 256, 0, stream>>>(h, W2, z);
    gat_attn2<<<(NN + 255) / 256, 256, 0, stream>>>(z, asc2, ads2, as2, ad2);
    gat_emax2<<<(EP + 255) / 256, 256, 0, stream>>>(ei, as2, ad2, m2);
    gat_esoft2<<<((long)EP * 16 + 255) / 256, 256, 0, stream>>>(ei, as2, ad2, m2, z, den2, acc2);
    gat_final<<<(nO + 255) / 256, 256, 0, stream>>>(acc2, den2, b2, out);
}